// YoloLoss_31585189495023
// MI455X (gfx1250) — compile-verified
//
#include <hip/hip_runtime.h>
#include <math.h>

#define NUM_CLASSES 80
#define AA 3            // anchors per scale
#define NCH 85          // 5 + NUM_CLASSES
#define BB 16           // batch
#define MM 50           // labels per image
#define EPSF 1e-7f
#define NPART 512

typedef __attribute__((ext_vector_type(2))) float v2f;
typedef __attribute__((ext_vector_type(8))) float v8f;

__constant__ float c_anchors[9][2] = {
  {12.f,16.f},{19.f,36.f},{40.f,28.f},{36.f,75.f},{76.f,55.f},
  {72.f,146.f},{142.f,110.f},{192.f,243.f},{459.f,401.f}
};

__device__ __forceinline__ float sigmoidf_(float x) {
  return 1.f / (1.f + expf(-x));
}

// CIoU between [0,0,w1,h1] and [0,0,w2,h2] (xyxy branch of bboxes_iou)
__device__ __forceinline__ float ciou_wh(float w1, float h1, float w2, float h2) {
  float iw = fminf(w1, w2), ih = fminf(h1, h2);
  float en = (iw > 0.f && ih > 0.f) ? 1.f : 0.f;
  float area_i = iw * ih * en;
  float area_u = w1 * h1 + w2 * h2 - area_i;
  float iou = area_i / area_u;
  float mw = fmaxf(w1, w2), mh = fmaxf(h1, h2);
  float c2 = mw * mw + mh * mh + 1e-16f;
  float dw = w1 - w2, dh = h1 - h2;
  float rho2 = dw * dw * 0.25f + dh * dh * 0.25f;
  float dat = atanf(w1 / h1) - atanf(w2 / h2);
  const float k4pi2 = 0.40528473456935108577f; // 4/pi^2
  float v = k4pi2 * dat * dat;
  float alpha = v / (1.f - iou + v);            // stop_gradient irrelevant in fwd
  return iou - (rho2 / c2 + v * alpha);
}

// ---- init: conf_mask = 1, tgt_idx = -1, partials = 0 ----------------------
__global__ __launch_bounds__(256)
void init_kernel(float* __restrict__ conf, int* __restrict__ tgt,
                 float* __restrict__ partials, int n) {
  int i = blockIdx.x * blockDim.x + threadIdx.x;
  int stride = gridDim.x * blockDim.x;
  for (int k = i; k < n; k += stride) { conf[k] = 1.f; tgt[k] = -1; }
  if (i < NPART) partials[i] = 0.f;
}

// ---- build_target: one wave per batch image -------------------------------
__global__ __launch_bounds__(32)
void build_kernel(const float* __restrict__ labels, float* __restrict__ conf,
                  int* __restrict__ tgt, float* __restrict__ lv,
                  int F, float stride, int m0) {
  __shared__ int s_sel[MM], s_bn[MM], s_ti[MM], s_tj[MM];
  int b = blockIdx.x;

  // Phase A: per-label quantities + ignore-zero scatter (idempotent 0-writes)
  for (int m = threadIdx.x; m < MM; m += 32) {
    const float* L = labels + (size_t)(b * MM + m) * 5;
    float X1 = L[0], Y1 = L[1], X2 = L[2], Y2 = L[3], C = L[4];
    bool valid = (X1 + Y1 + X2 + Y2 + C) > 0.f;
    float tx = (X2 + X1) / (2.f * stride);
    float ty = (Y2 + Y1) / (2.f * stride);
    float tw = (X2 - X1) / stride;
    float th = (Y2 - Y1) / stride;
    int ti = (int)tx, tj = (int)ty;

    float best = -1e30f; int besti = 0;
    float ioum[AA];
    for (int k = 0; k < 9; k++) {
      float w2 = c_anchors[k][0] / stride, h2 = c_anchors[k][1] / stride;
      float io = ciou_wh(tw, th, w2, h2);
      if (io > best) { best = io; besti = k; }   // first-max wins, like argmax
      if (k >= m0 && k < m0 + AA) ioum[k - m0] = io;
    }
    int bn = besti % AA;
    bool sel = valid && (besti >= m0) && (besti < m0 + AA);

    if (sel) {
      for (int a = 0; a < AA; a++) {
        // reference scatters zeros at [b, a, ti, tj] (note transposed vs target!)
        if (ioum[a] > 0.5f && ti >= 0 && ti < F && tj >= 0 && tj < F)
          conf[(((size_t)b * AA + a) * F + ti) * F + tj] = 0.f;
      }
    }
    float fx = tx - (float)ti, fy = ty - (float)tj;
    float mw = c_anchors[m0 + bn][0] / stride;
    float mh = c_anchors[m0 + bn][1] / stride;
    float* V = lv + (size_t)(b * MM + m) * 8;
    V[0] = fx; V[1] = fy;
    V[2] = logf(tw / mw + 1e-16f);
    V[3] = logf(th / mh + 1e-16f);
    V[4] = C;
    s_sel[m] = sel ? 1 : 0; s_bn[m] = bn; s_ti[m] = ti; s_tj[m] = tj;
  }
  __syncthreads();

  // Phase B: serial 1-scatter (all zeros happen before any ones; last label wins)
  if (threadIdx.x == 0) {
    for (int m = 0; m < MM; m++) {
      if (s_sel[m]) {
        int a = s_bn[m], ti = s_ti[m], tj = s_tj[m];
        if (ti >= 0 && ti < F && tj >= 0 && tj < F) {
          size_t idx = (((size_t)b * AA + a) * F + tj) * F + ti; // [b, bn, tj, ti]
          conf[idx] = 1.f;
          tgt[idx] = b * MM + m;
        }
      }
    }
  }
}

// ---- deterministic wave-level WMMA sum of nvals floats (multiple of 64) ------------------
// A = ones(16x4), B = 64 values/chunk, C accumulates. Every B element lands in
// all 16 rows of its column => sum(all 256 C entries) == 16 * sum(values).
__device__ __forceinline__ void wmma_reduce_write(const float* sdata, int nvals,
                                                  float* swave, float* dst) {
  v2f av; av.x = 1.f; av.y = 1.f;
  v8f cacc = {};
  for (int ch = 0; ch < nvals; ch += 64) {
    v2f bv;
    bv.x = sdata[ch + (int)threadIdx.x * 2];
    bv.y = sdata[ch + (int)threadIdx.x * 2 + 1];
    cacc = __builtin_amdgcn_wmma_f32_16x16x4_f32(
        false, av, false, bv, (short)0, cacc, false, false);
  }
  float s8 = cacc[0] + cacc[1] + cacc[2] + cacc[3] +
             cacc[4] + cacc[5] + cacc[6] + cacc[7];
  swave[threadIdx.x] = s8;
  asm volatile("s_wait_dscnt 0" ::: "memory");   // order in-wave cross-lane LDS
  if (threadIdx.x == 0) {
    float tot = 0.f;
    for (int l = 0; l < 32; l++) tot += swave[l];
    *dst = tot * 0.0625f;                        // /16, exact in fp32
  }
}

// ---- main loss: only conf channel everywhere + 85ch at rare target cells --
__global__ __launch_bounds__(256)
void loss_kernel(const float* __restrict__ x, const float* __restrict__ conf,
                 const int* __restrict__ tgt, const float* __restrict__ lv,
                 float* __restrict__ partials, int poff, int F) {
  const int FF = F * F;
  const int N = BB * AA * FF;
  float acc = 0.f;
  for (int c = blockIdx.x * blockDim.x + threadIdx.x; c < N;
       c += gridDim.x * blockDim.x) {
    int i = c % F;
    int j = (c / F) % F;
    int a = (c / FF) % AA;
    int b = c / (FF * AA);
    const float* xb = x + (size_t)(b * (AA * NCH) + a * NCH) * FF + j * F + i;

    float p  = sigmoidf_(xb[4 * FF]);
    float pc = fminf(fmaxf(p, EPSF), 1.f - EPSF);
    int   t  = tgt[c];
    if (t < 0) {
      // noobj: L_NOOBJ * conf_mask * BCE(p, 0)
      acc += conf[c] * (-log1pf(-pc));
    } else {
      // obj: L_OBJ * BCE(p, 1)
      acc += 10.0f * (-logf(pc));
      const float* V = lv + (size_t)t * 8;
      float d0 = sigmoidf_(xb[0])      - V[0];
      float d1 = sigmoidf_(xb[FF])     - V[1];
      float d2 = xb[2 * FF]            - V[2];
      float d3 = xb[3 * FF]            - V[3];
      acc += d0 * d0 + d1 * d1 + d2 * d2 + d3 * d3;  // L_COORD = 1
      int cid = (int)V[4];
      for (int cc = 0; cc < NUM_CLASSES; cc++) {
        float pl  = sigmoidf_(xb[(5 + cc) * FF]);
        float plc = fminf(fmaxf(pl, EPSF), 1.f - EPSF);
        acc += (cc == cid) ? (-logf(plc)) : (-log1pf(-plc));
      }
    }
  }

  __shared__ float sdata[256];
  __shared__ float swave[32];
  sdata[threadIdx.x] = acc;
  __syncthreads();
  if (threadIdx.x < 32) {  // wave 0, full EXEC
    wmma_reduce_write(sdata, 256, swave, &partials[poff + blockIdx.x]);
  }
}

// ---- final deterministic reduction of all block partials ------------------
__global__ __launch_bounds__(32)
void final_kernel(const float* __restrict__ partials, float* __restrict__ out) {
  __shared__ float swave[32];
  wmma_reduce_write(partials, NPART, swave, out);
}

extern "C" void kernel_launch(void* const* d_in, const int* in_sizes, int n_in,
                              void* d_out, int out_size, void* d_ws, size_t ws_size,
                              hipStream_t stream) {
  (void)in_sizes; (void)n_in; (void)out_size;
  const float* x0     = (const float*)d_in[0];
  const float* x1     = (const float*)d_in[1];
  const float* x2     = (const float*)d_in[2];
  const float* labels = (const float*)d_in[3];
  float* out = (float*)d_out;

  const size_t S0 = (size_t)BB * AA * 76 * 76;   // 277248
  const size_t S1 = (size_t)BB * AA * 38 * 38;   //  69312
  const size_t S2 = (size_t)BB * AA * 19 * 19;   //  17328
  const size_t ST = S0 + S1 + S2;                // 363888
  const size_t LVS = (size_t)BB * MM * 8;        //   6400 per scale

  float* conf     = (float*)d_ws;                // ST floats
  int*   tgt      = (int*)(conf + ST);           // ST ints
  float* lv       = (float*)(tgt + ST);          // 3*LVS floats
  float* partials = lv + 3 * LVS;                // NPART floats
  if (ws_size < (ST * 2 + 3 * LVS + NPART) * 4) return;

  int initBlocks = (int)((ST + 255) / 256);
  init_kernel<<<initBlocks, 256, 0, stream>>>(conf, tgt, partials, (int)ST);

  build_kernel<<<BB, 32, 0, stream>>>(labels, conf,           tgt,           lv,           76,  8.f, 0);
  build_kernel<<<BB, 32, 0, stream>>>(labels, conf + S0,      tgt + S0,      lv + LVS,     38, 16.f, 3);
  build_kernel<<<BB, 32, 0, stream>>>(labels, conf + S0 + S1, tgt + S0 + S1, lv + 2 * LVS, 19, 32.f, 6);

  const int NB0 = 256, NB1 = 64, NB2 = 32;       // 352 partials <= NPART
  loss_kernel<<<NB0, 256, 0, stream>>>(x0, conf,           tgt,           lv,           partials, 0,         76);
  loss_kernel<<<NB1, 256, 0, stream>>>(x1, conf + S0,      tgt + S0,      lv + LVS,     partials, NB0,       38);
  loss_kernel<<<NB2, 256, 0, stream>>>(x2, conf + S0 + S1, tgt + S0 + S1, lv + 2 * LVS, partials, NB0 + NB1, 19);

  final_kernel<<<1, 32, 0, stream>>>(partials, out);
}